// RESCALVAE_5308579578123
// MI455X (gfx1250) — compile-verified
//
#include <hip/hip_runtime.h>
#include <hip/hip_bf16.h>

// Problem constants (match reference)
#define BB 4
#define NN 512
#define DD 64
#define EE 16384
#define RR 512
#define NEDGE (BB * EE)          // 65536
#define LDS_STRIDE 68            // 64 + 4 pad: kills LDS bank conflicts on row-major A-frag reads

typedef __attribute__((ext_vector_type(2))) float v2f;
typedef __attribute__((ext_vector_type(8))) float v8f;

// ---------------------------------------------------------------------------
// Pass 1: histogram of relation ids; invalid relations get score 0 directly.
// ---------------------------------------------------------------------------
__global__ void rescal_count_kernel(const long long* __restrict__ edges,
                                    int* __restrict__ counts,
                                    float* __restrict__ out) {
    int tid = blockIdx.x * blockDim.x + threadIdx.x;   // tid == b*E + e
    long long r = edges[(long long)tid * 3 + 1];
    if (r >= 0 && r < RR) {
        atomicAdd(&counts[(int)r], 1);
    } else {
        out[tid] = 0.0f;                               // jnp.where(valid, ..., 0)
    }
}

// ---------------------------------------------------------------------------
// Pass 2: exclusive scan over 512 counts (single 512-thread block).
// ---------------------------------------------------------------------------
__global__ void rescal_scan_kernel(const int* __restrict__ counts,
                                   int* __restrict__ offsets,
                                   int* __restrict__ cursors) {
    __shared__ int tmp[RR];
    int t = threadIdx.x;
    int c = counts[t];
    tmp[t] = c;
    __syncthreads();
    for (int off = 1; off < RR; off <<= 1) {
        int v = (t >= off) ? tmp[t - off] : 0;
        __syncthreads();
        tmp[t] += v;
        __syncthreads();
    }
    int excl = tmp[t] - c;
    offsets[t] = excl;
    cursors[t] = excl;
}

// ---------------------------------------------------------------------------
// Pass 3: scatter edge ids into relation buckets. Order within a bucket is
// non-deterministic, but every edge's score is computed column-independently
// and written to its own slot, so the OUTPUT is bitwise deterministic.
// ---------------------------------------------------------------------------
__global__ void rescal_scatter_kernel(const long long* __restrict__ edges,
                                      int* __restrict__ cursors,
                                      int* __restrict__ eids) {
    int tid = blockIdx.x * blockDim.x + threadIdx.x;
    long long r = edges[(long long)tid * 3 + 1];
    if (r >= 0 && r < RR) {
        int pos = atomicAdd(&cursors[(int)r], 1);
        eids[pos] = tid;
    }
}

// ---------------------------------------------------------------------------
// Pass 4: per relation r: stage R_r (64x64) in LDS, then for tiles of 16
// edges compute T = R * O  (64x16) with v_wmma_f32_16x16x4_f32 and reduce
// score_j = s_j . T[:,j] directly from the C/D fragment layout.
//
// Fragment layouts (ISA 7.12.2, wave32):
//   A (16x4 f32, 2 VGPRs): lanes 0-15 -> M=lane, K={0,1}; lanes 16-31 -> K={2,3}
//   B (4x16 f32, 2 VGPRs): lanes 0-15 -> N=lane, K={0,1}; lanes 16-31 -> K={2,3}
//   C/D (16x16 f32, 8 VGPRs): lane&15 = column N; v = row (lanes>=16: row v+8)
// ---------------------------------------------------------------------------
__global__ __launch_bounds__(128) void rescal_wmma_kernel(
    const float* __restrict__ node_emb,
    const float* __restrict__ relmat,
    const long long* __restrict__ edges,
    const int* __restrict__ offsets,
    const int* __restrict__ counts,
    const int* __restrict__ eids,
    float* __restrict__ out) {

    __shared__ float ldsR[DD * LDS_STRIDE];

    const int r = blockIdx.x;
    const float* Rg = relmat + (size_t)r * (DD * DD);
    for (int i = threadIdx.x; i < DD * DD; i += 128) {
        ldsR[(i >> 6) * LDS_STRIDE + (i & 63)] = Rg[i];
    }
    __syncthreads();

    const int cnt = counts[r];
    if (cnt == 0) return;                 // uniform per workgroup
    const int base = offsets[r];

    const int lane = threadIdx.x & 31;
    const int wave = threadIdx.x >> 5;
    const int col  = lane & 15;
    const bool hi  = lane >= 16;
    const int koff = hi ? 2 : 0;          // K sub-pair owned by this half-wave

    // waves stride over 16-edge tiles; loop bound uniform across the wave
    for (int tbase = wave * 16; tbase < cnt; tbase += 16 * 4) {
        const int rem   = cnt - tbase;                  // >= 1 here
        const bool valid = col < rem;
        const int slot  = valid ? col : 0;              // clamp tail lanes to a real edge
        const int gid   = eids[base + tbase + slot];
        const int b     = gid >> 14;                    // gid / E (E = 16384)

        const long long* ep = edges + (long long)gid * 3;
        long long si = ep[0], oi = ep[2];
        si = si < 0 ? 0 : (si > NN - 1 ? NN - 1 : si);  // jnp.clip
        oi = oi < 0 ? 0 : (oi > NN - 1 ? NN - 1 : oi);
        const float* sp = node_emb + ((size_t)b * NN + (int)si) * DD;
        const float* op = node_emb + ((size_t)b * NN + (int)oi) * DD;

        // Preload all 16 B fragments: this lane's half of its edge's o-vector.
        v2f bf[16];
#pragma unroll
        for (int kc = 0; kc < 16; ++kc) {
            bf[kc] = *(const v2f*)(op + kc * 4 + koff);
        }

        float partial = 0.0f;
#pragma unroll
        for (int mb = 0; mb < 4; ++mb) {
            v8f acc = {};
            const int row = mb * 16 + col;              // A row owned by this lane
#pragma unroll
            for (int kc = 0; kc < 16; ++kc) {
                v2f af = *(const v2f*)(&ldsR[row * LDS_STRIDE + kc * 4 + koff]);
                acc = __builtin_amdgcn_wmma_f32_16x16x4_f32(
                    /*neg_a=*/false, af, /*neg_b=*/false, bf[kc],
                    /*c_mod=*/(short)0, acc, /*reuse_a=*/false, /*reuse_b=*/false);
            }
            // score partial: lanes 0-15 own rows mb*16+0..7 of column `col`,
            // lanes 16-31 own rows mb*16+8..15 of the same column.
            const float* sblk = sp + mb * 16 + (hi ? 8 : 0);
#pragma unroll
            for (int v = 0; v < 8; ++v) partial += sblk[v] * acc[v];
        }

        // combine the two half-rows of each column
        partial += __shfl_xor(partial, 16, 32);
        if (!hi && valid) out[gid] = partial;
    }
}

// ---------------------------------------------------------------------------
extern "C" void kernel_launch(void* const* d_in, const int* in_sizes, int n_in,
                              void* d_out, int out_size, void* d_ws, size_t ws_size,
                              hipStream_t stream) {
    const float*     node_emb = (const float*)d_in[0];      // (B,N,D) f32
    const float*     relmat   = (const float*)d_in[1];      // (R,D,D) f32
    const long long* edges    = (const long long*)d_in[2];  // (B,E,3) i64
    float* out = (float*)d_out;                             // (B,E) f32

    char* ws = (char*)d_ws;
    int* counts  = (int*)(ws);                // 512 ints
    int* offsets = (int*)(ws + 2048);         // 512 ints
    int* cursors = (int*)(ws + 4096);         // 512 ints
    int* eids    = (int*)(ws + 6144);         // 65536 ints

    hipMemsetAsync(counts, 0, RR * sizeof(int), stream);
    rescal_count_kernel<<<NEDGE / 256, 256, 0, stream>>>(edges, counts, out);
    rescal_scan_kernel<<<1, RR, 0, stream>>>(counts, offsets, cursors);
    rescal_scatter_kernel<<<NEDGE / 256, 256, 0, stream>>>(edges, cursors, eids);
    rescal_wmma_kernel<<<RR, 128, 0, stream>>>(node_emb, relmat, edges,
                                               offsets, counts, eids, out);
}